// KroneckerLinear_52364241273353
// MI455X (gfx1250) — compile-verified
//
#include <hip/hip_runtime.h>

typedef __attribute__((ext_vector_type(2))) float v2f;
typedef __attribute__((ext_vector_type(4))) float v4f;
typedef __attribute__((ext_vector_type(8))) float v8f;
typedef __attribute__((ext_vector_type(4))) int   v4i;

#define DIM    128
#define PITCH  132   // padded LDS pitch (floats): 132 % 64 == 4 -> conflict-free strided access
#define TOKENS 2048
#define GRID   512

typedef __attribute__((address_space(1))) v4i g_v4i;   // global 128-bit chunk
typedef __attribute__((address_space(3))) v4i l_v4i;   // LDS 128-bit chunk

// ---- CDNA5 async global->LDS copy (ASYNCcnt path) ---------------------------
__device__ __forceinline__ void async_b128_to_lds(const float* src, float* dst_lds) {
    // LDS flat pointers carry the byte offset in addr[31:0] (ISA 10.2 aperture rules)
    unsigned dst_off = (unsigned)(unsigned long long)dst_lds;
#if __has_builtin(__builtin_amdgcn_global_load_async_to_lds_b128)
    __builtin_amdgcn_global_load_async_to_lds_b128(
        (g_v4i*)(unsigned long long)src,
        (l_v4i*)dst_off, 0, 0);
#else
    asm volatile("global_load_async_to_lds_b128 %0, %1, off"
                 :: "v"(dst_off), "v"(src)
                 : "memory");
#endif
}

__device__ __forceinline__ void wait_async0() {
#if __has_builtin(__builtin_amdgcn_s_wait_asynccnt)
    __builtin_amdgcn_s_wait_asynccnt(0);
#else
    asm volatile("s_wait_asynccnt 0x0" ::: "memory");
#endif
}

// Each wave stages 16 rows of X_t (one full 512B row per b128 instruction).
__device__ __forceinline__ void stage_x_async(const float* xt, float* Xs,
                                              int wave, int lane) {
#pragma unroll
    for (int i = 0; i < 16; ++i) {
        const int row = wave * 16 + i;
        async_b128_to_lds(xt + row * DIM + lane * 4,
                          Xs + row * PITCH + lane * 4);
    }
}

// y[t, j1*128+j2] = sum_{i1,i2} x[t, i1*128+i2] * U[i1,j1] * V[i2,j2]
//   Z_t = X_t @ V ;  Y_t = U^T @ Z_t
__global__ __launch_bounds__(256) void kron_wmma_f32(
    const float* __restrict__ x,   // [2048][16384]
    const float* __restrict__ U,   // [128][128]
    const float* __restrict__ V,   // [128][128]
    float* __restrict__ y,         // [2048][16384]
    int tokens_per_wg)
{
    __shared__ float Xs[DIM * PITCH];   // 66 KB, X_t row-major (padded)
    __shared__ float Zt[DIM * PITCH];   // 66 KB, Z_t TRANSPOSED: Zt[col][row]

    const int tid  = threadIdx.x;
    const int lane = tid & 31;
    const int wave = tid >> 5;          // 0..7
    const int half = lane >> 4;         // lanes 16..31 hold K+2 in A/B fragments
    const int l16  = lane & 15;
    const int colw = 16 * wave + l16;

    // ---- stationary fragments in registers ---------------------------------
    // vfrag[kk]: B-frag of V (K rows 4kk..4kk+3, cols 16w..16w+15)
    // ufrag[kk]: A-frag of U^T (rows 16w..16w+15): A[m][k] = U[k][16w+m]
    v2f vfrag[32];
    v2f ufrag[32];
#pragma unroll
    for (int kk = 0; kk < 32; ++kk) {
        const int k0 = 4 * kk + 2 * half;
        vfrag[kk][0] = V[(k0 + 0) * DIM + colw];
        vfrag[kk][1] = V[(k0 + 1) * DIM + colw];
        ufrag[kk][0] = U[(k0 + 0) * DIM + colw];
        ufrag[kk][1] = U[(k0 + 1) * DIM + colw];
    }

    const long t0 = (long)blockIdx.x * tokens_per_wg;

    // ---- prologue: async-stage first token ----------------------------------
    stage_x_async(x + t0 * (long)(DIM * DIM), Xs, wave, lane);
    wait_async0();
    __syncthreads();

    for (int ti = 0; ti < tokens_per_wg; ++ti) {
        const long t = t0 + ti;

        // ---- mm1: Z = X_t @ V ; wave w -> col block w, 8 row blocks ---------
        for (int r = 0; r < 8; ++r) {
            v8f acc = {0.f, 0.f, 0.f, 0.f, 0.f, 0.f, 0.f, 0.f};
            const float* __restrict__ xrow = Xs + (16 * r + l16) * PITCH + 2 * half;
#pragma unroll
            for (int kk = 0; kk < 32; ++kk) {
                v2f a;
                a[0] = xrow[4 * kk + 0];
                a[1] = xrow[4 * kk + 1];
                acc = __builtin_amdgcn_wmma_f32_16x16x4_f32(
                    false, a, false, vfrag[kk], (short)0, acc, false, false);
            }
            // acc[v] -> Z row 16r+8*half+v, col colw ; store transposed, b128 x2
            float* __restrict__ zout = Zt + colw * PITCH + 16 * r + 8 * half;
            v4f lo = {acc[0], acc[1], acc[2], acc[3]};
            v4f hi = {acc[4], acc[5], acc[6], acc[7]};
            *(v4f*)(zout)     = lo;
            *(v4f*)(zout + 4) = hi;
        }
        __syncthreads();   // all Xs reads done; Zt complete

        // ---- overlap: prefetch next token's X while mm2 runs ----------------
        if (ti + 1 < tokens_per_wg)
            stage_x_async(x + (t + 1) * (long)(DIM * DIM), Xs, wave, lane);

        // ---- mm2: Y_t = U^T @ Z ; wave w -> row block w, 8 col blocks -------
        float* __restrict__ yt = y + t * (long)(DIM * DIM);
        for (int cb = 0; cb < 8; ++cb) {
            v8f acc = {0.f, 0.f, 0.f, 0.f, 0.f, 0.f, 0.f, 0.f};
            // B[k][n] = Z[k][16cb+n] = Zt[16cb+n][k]; consecutive k -> contiguous
            const float* __restrict__ zcol = Zt + (16 * cb + l16) * PITCH + 2 * half;
#pragma unroll
            for (int kk = 0; kk < 32; ++kk) {
                v2f b;
                b[0] = zcol[4 * kk + 0];
                b[1] = zcol[4 * kk + 1];
                acc = __builtin_amdgcn_wmma_f32_16x16x4_f32(
                    false, ufrag[kk], false, b, (short)0, acc, false, false);
            }
            float* __restrict__ yo = yt + (16 * wave + 8 * half) * DIM + 16 * cb + l16;
#pragma unroll
            for (int v = 0; v < 8; ++v)
                __builtin_nontemporal_store(acc[v], yo + v * DIM);
        }

        wait_async0();     // next X tile resident in LDS
        __syncthreads();   // all Zt reads done; Xs refilled for next iteration
    }
}

extern "C" void kernel_launch(void* const* d_in, const int* in_sizes, int n_in,
                              void* d_out, int out_size, void* d_ws, size_t ws_size,
                              hipStream_t stream) {
    const float* x = (const float*)d_in[0];
    const float* U = (const float*)d_in[1];
    const float* V = (const float*)d_in[2];
    float* y = (float*)d_out;
    (void)in_sizes; (void)n_in; (void)out_size; (void)d_ws; (void)ws_size;

    const int tokens_per_wg = TOKENS / GRID;   // 4
    kron_wmma_f32<<<GRID, 256, 0, stream>>>(x, U, V, y, tokens_per_wg);
}